// ConfidenceBCELoss_9672266350835
// MI455X (gfx1250) — compile-verified
//
#include <hip/hip_runtime.h>
#include <stdint.h>

// ---------------- problem constants ----------------
#define Bdim 2048
#define Sdim 16384
#define DELAYF 5.0f

// ---------------- tiling ----------------
#define ROWS_PER_TILE 16                 // WMMA M
#define TILES_B (Bdim / ROWS_PER_TILE)   // 128
#define CHUNK 512                        // columns per block
#define COL_SPLITS (Sdim / CHUNK)        // 32
#define PITCH 516                        // 512 + 4 pad DWORDs (TDM pad: 1dw / 128dw)
#define THREADS 256
#define WAVES (THREADS / 32)

typedef __attribute__((ext_vector_type(2))) float v2f;
typedef __attribute__((ext_vector_type(8))) float v8f;
typedef __attribute__((ext_vector_type(4))) unsigned int v4u;
typedef __attribute__((ext_vector_type(4))) int v4i;
typedef __attribute__((ext_vector_type(8))) int v8i;

#ifndef __has_builtin
#define __has_builtin(x) 0
#endif

#if __has_builtin(__builtin_amdgcn_wmma_f32_16x16x4_f32)
#define HAVE_WMMA 1
#else
#define HAVE_WMMA 0
#endif

#if __has_builtin(__builtin_amdgcn_tensor_load_to_lds) && \
    __has_builtin(__builtin_amdgcn_s_wait_tensorcnt)
#define HAVE_TDM 1
#else
#define HAVE_TDM 0
#endif

// Raw hardware transcendentals: v_exp_f32 / v_log_f32 are base-2.
// Our log argument is 1+e in (1,2] (never denorm/inf), so no fixups needed.
#if __has_builtin(__builtin_amdgcn_exp2f) && __has_builtin(__builtin_amdgcn_logf)
#define EXP2F(v) __builtin_amdgcn_exp2f(v)
#define LOG2F(v) __builtin_amdgcn_logf(v)
#else
#define EXP2F(v) exp2f(v)
#define LOG2F(v) log2f(v)
#endif
#define LOG2E_F 1.44269504088896340736f
#define LN2_F 0.69314718055994530942f

// ============================================================
// Kernel 1: last-nonzero index per row. O(1) expected work/row:
// region >= lengths[r] is zero-padded, so scan back from lengths-1.
// Reference: idx = S-1-argmax(reversed nz); all-zero row -> S-1.
// ============================================================
__global__ __launch_bounds__(256) void last_nz_kernel(
    const float* __restrict__ x, const int* __restrict__ len,
    int* __restrict__ idx) {
  int r = blockIdx.x * blockDim.x + threadIdx.x;
  if (r >= Bdim) return;
  int j = len[r] - 1;
  if (j > Sdim - 1) j = Sdim - 1;
  if (j < 0) j = 0;
  const float* xr = x + (size_t)r * Sdim;
  while (j > 0 && xr[j] == 0.0f) --j;
  idx[r] = (xr[j] != 0.0f) ? j : (Sdim - 1);
}

// ============================================================
// TDM: DMA a 16 x CHUNK f32 tile (row stride Sdim) into LDS with
// 1-DWORD padding every 128 DWORDs (pitch 516 -> conflict-free reads).
// ============================================================
#if HAVE_TDM
__device__ __forceinline__ void tdm_load_tile(const float* gsrc,
                                              unsigned lds_byte_off) {
  uint64_t ga = (uint64_t)(uintptr_t)gsrc;
  // ---- D# group 0 (128b): count=1 | lds_addr | global_addr[56:0] | type=2
  v4u g0;
  g0[0] = 1u;
  g0[1] = lds_byte_off;
  g0[2] = (unsigned)(ga & 0xFFFFFFFFu);
  g0[3] = (unsigned)((ga >> 32) & 0x01FFFFFFu) | (2u << 30);
  // ---- D# group 1 (256b)
  // w0: wg_mask=0 | data_size=2(4B)<<16 | pad_enable<<20 | pad_interval=6(128dw)<<22 | pad_amount=0(1dw)
  const unsigned td0 = 0x40000000u;  // generous dims: tiles always in-bounds
  const unsigned td1 = 0x40000000u;
  v8i g1;
  g1[0] = (int)((2u << 16) | (1u << 20) | (6u << 22));
  g1[1] = (int)((td0 & 0xFFFFu) << 16);                       // tensor_dim0 lo16
  g1[2] = (int)((td0 >> 16) | ((td1 & 0xFFFFu) << 16));       // dim0 hi | dim1 lo
  g1[3] = (int)((td1 >> 16) | ((unsigned)CHUNK << 16));       // dim1 hi | tile_dim0
  g1[4] = (int)(ROWS_PER_TILE & 0xFFFF);                      // tile_dim1=16, tile_dim2=0
  g1[5] = (int)Sdim;                                          // tensor_dim0_stride lo32
  g1[6] = 0;                                                  // stride hi16 | dim1_stride lo16
  g1[7] = 0;
  v4i gz = {0, 0, 0, 0};
#if defined(__clang_major__) && (__clang_major__ >= 23)
  v8i gz8 = {0, 0, 0, 0, 0, 0, 0, 0};
  __builtin_amdgcn_tensor_load_to_lds(g0, g1, gz, gz, gz8, 0);
#else
  __builtin_amdgcn_tensor_load_to_lds(g0, g1, gz, gz, 0);
#endif
}
#endif

// ============================================================
// Kernel 2: per-tile weighted BCE sums.
// Block = 16 rows x 512 cols; 8 waves each own a 64-col slice.
// Elementwise bce on VALU; row-sum accumulation offloaded to the
// matrix pipe: D += A(16x4) * ones(4x16) via V_WMMA_F32_16X16X4_F32.
// ============================================================
__global__ __launch_bounds__(THREADS) void tile_loss_kernel(
    const float* __restrict__ x, const float* __restrict__ target,
    const int* __restrict__ idx, float* __restrict__ pn,
    float* __restrict__ pd) {
  __shared__ float tile[ROWS_PER_TILE * PITCH];  // padded f32 tile
  __shared__ float wbuf[CHUNK];                  // sigmoid(j - 5) LUT
  __shared__ float rn[WAVES][ROWS_PER_TILE];
  __shared__ float rd[WAVES][ROWS_PER_TILE];

  const int tid = threadIdx.x;
  const int lane = tid & 31;
  const int wv = tid >> 5;
  const int row = lane & 15;   // WMMA A: lane -> M
  const int half = lane >> 4;  // lanes 16-31 carry K=2,3
  const int r0 = blockIdx.x * ROWS_PER_TILE;
  const int c0 = blockIdx.y * CHUNK;

#if HAVE_TDM
  if (tid < 32) {  // one wave issues the DMA (EXEC ignored by TDM)
    tdm_load_tile(x + ((size_t)r0 * Sdim + c0), (unsigned)(uintptr_t)&tile[0]);
  }
#else
  for (int e = tid; e < ROWS_PER_TILE * CHUNK; e += THREADS) {
    int r = e / CHUNK;
    int j = e % CHUNK;
    tile[r * PITCH + j + (j >> 7)] = x[(size_t)(r0 + r) * Sdim + c0 + j];
  }
#endif

  // column weight LUT while DMA is in flight
  for (int j = tid; j < CHUNK; j += THREADS) {
    float p = (float)(c0 + j) - DELAYF;
    wbuf[j] = 1.0f / (1.0f + EXP2F(-p * LOG2E_F));
  }

  const int rowg = r0 + row;
  const float t = target[rowg];
  const int cut = idx[rowg];

#if HAVE_TDM
  if (tid < 32) __builtin_amdgcn_s_wait_tensorcnt(0);
#endif
  __syncthreads();

  v8f cn = {};  // num accumulator (16x16 C, row sums)
  v8f cd = {};  // den accumulator
  v2f bones;
  bones.x = 1.0f;
  bones.y = 1.0f;

  const int jbase = wv * (CHUNK / WAVES) + half * 2;
  const float* trow = &tile[row * PITCH];
#pragma unroll 4
  for (int it = 0; it < (CHUNK / WAVES) / 4; ++it) {
    int j = jbase + it * 4;
    int o = j + (j >> 7);  // padded offset; j even, j+1 never crosses pad
    float x0 = trow[o];
    float x1 = trow[o + 1];
    // Unconditional 8B-aligned weight-pair load (wbuf base aligned, j even),
    // then mask via value-select: keeps EXEC all-ones, no predicated loads.
    float2 wp = *(const float2*)(wbuf + j);
    int gj0 = c0 + j;
    float w0 = (gj0 < cut) ? wp.x : 0.0f;
    float w1 = (gj0 + 1 < cut) ? wp.y : 0.0f;
    // bce = max(x,0) - x*t + log1p(exp(-|x|)); raw exp2/log2, arg in (1,2]
    float e0 = EXP2F(fabsf(x0) * -LOG2E_F);
    float e1 = EXP2F(fabsf(x1) * -LOG2E_F);
    float l0 = LOG2F(1.0f + e0) * LN2_F;
    float l1 = LOG2F(1.0f + e1) * LN2_F;
    float b0 = fmaxf(x0, 0.0f) - x0 * t + l0;
    float b1 = fmaxf(x1, 0.0f) - x1 * t + l1;
    v2f an, ad;
    an.x = b0 * w0;
    an.y = b1 * w1;
    ad.x = w0;
    ad.y = w1;
#if HAVE_WMMA
    cn = __builtin_amdgcn_wmma_f32_16x16x4_f32(false, an, false, bones,
                                               (short)0, cn, false, false);
    cd = __builtin_amdgcn_wmma_f32_16x16x4_f32(false, ad, false, bones,
                                               (short)0, cd, false, false);
#else
    cn.s0 += an.x + an.y;
    cd.s0 += ad.x + ad.y;
#endif
  }

#if HAVE_WMMA
  // C/D layout: VGPR v, lanes 0-15 -> M=v; lanes 16-31 -> M=v+8 (all N equal)
  if (lane == 0) {
    for (int v = 0; v < 8; ++v) { rn[wv][v] = cn[v]; rd[wv][v] = cd[v]; }
  } else if (lane == 16) {
    for (int v = 0; v < 8; ++v) { rn[wv][8 + v] = cn[v]; rd[wv][8 + v] = cd[v]; }
  }
  __syncthreads();
  if (tid < ROWS_PER_TILE) {
    float sn = 0.0f, sd = 0.0f;
    for (int w = 0; w < WAVES; ++w) { sn += rn[w][tid]; sd += rd[w][tid]; }
    size_t rw = (size_t)(r0 + tid);
    pn[rw * COL_SPLITS + blockIdx.y] = sn;
    pd[rw * COL_SPLITS + blockIdx.y] = sd;
  }
#else
  __shared__ float lnum[WAVES][32];
  __shared__ float lden[WAVES][32];
  lnum[wv][lane] = cn.s0;
  lden[wv][lane] = cd.s0;
  __syncthreads();
  if (tid < ROWS_PER_TILE) {
    float sn = 0.0f, sd = 0.0f;
    for (int w = 0; w < WAVES; ++w) {
      sn += lnum[w][tid] + lnum[w][tid + 16];
      sd += lden[w][tid] + lden[w][tid + 16];
    }
    size_t rw = (size_t)(r0 + tid);
    pn[rw * COL_SPLITS + blockIdx.y] = sn;
    pd[rw * COL_SPLITS + blockIdx.y] = sd;
  }
#endif
}

// ============================================================
// Kernel 3: deterministic final reduce: mean_r( num_r / den_r ).
// ============================================================
__global__ __launch_bounds__(256) void finalize_kernel(
    const float* __restrict__ pn, const float* __restrict__ pd,
    float* __restrict__ out) {
  __shared__ float red[256];
  int tid = threadIdx.x;
  float acc = 0.0f;
  for (int r = tid; r < Bdim; r += 256) {
    float sn = 0.0f, sd = 0.0f;
    for (int s = 0; s < COL_SPLITS; ++s) {
      sn += pn[(size_t)r * COL_SPLITS + s];
      sd += pd[(size_t)r * COL_SPLITS + s];
    }
    acc += sn / sd;
  }
  red[tid] = acc;
  __syncthreads();
  for (int st = 128; st > 0; st >>= 1) {
    if (tid < st) red[tid] += red[tid + st];
    __syncthreads();
  }
  if (tid == 0) out[0] = red[0] / (float)Bdim;
}

// ============================================================
// Launcher. ws layout (floats): pn[B*32] | pd[B*32] | idx[B] (int)
//   = 2048*32*2*4 + 2048*4 = ~532 KB of workspace.
// ============================================================
extern "C" void kernel_launch(void* const* d_in, const int* in_sizes, int n_in,
                              void* d_out, int out_size, void* d_ws,
                              size_t ws_size, hipStream_t stream) {
  (void)in_sizes; (void)n_in; (void)out_size; (void)ws_size;
  const float* x = (const float*)d_in[0];
  const int* len = (const int*)d_in[1];
  const float* tgt = (const float*)d_in[2];
  float* out = (float*)d_out;

  float* ws_f = (float*)d_ws;
  float* pn = ws_f;
  float* pd = ws_f + (size_t)Bdim * COL_SPLITS;
  int* idx = (int*)(ws_f + 2 * (size_t)Bdim * COL_SPLITS);

  last_nz_kernel<<<(Bdim + 255) / 256, 256, 0, stream>>>(x, len, idx);

  dim3 grid(TILES_B, COL_SPLITS);
  tile_loss_kernel<<<grid, THREADS, 0, stream>>>(x, tgt, idx, pn, pd);

  finalize_kernel<<<1, 256, 0, stream>>>(pn, pd, out);
}